// Extra_layers_7722351199021
// MI455X (gfx1250) — compile-verified
//
#include <hip/hip_runtime.h>
#include <stdint.h>

typedef __attribute__((ext_vector_type(2))) float v2f;
typedef __attribute__((ext_vector_type(8))) float v8f;

#define BN_EPS 1e-5f

// ---------------------------------------------------------------------------
// CDNA5 helpers: async global->LDS DMA (ASYNCcnt path)
// ---------------------------------------------------------------------------
__device__ __forceinline__ uint32_t lds_off(const void* p) {
  // flat shared pointer = {shared_aperture_hi32, lds_offset_lo32}
  return (uint32_t)(uintptr_t)p;
}

__device__ __forceinline__ void async_ld_f32(uint32_t lds_dst, const float* src) {
  asm volatile("global_load_async_to_lds_b32 %0, %1, off"
               :: "v"(lds_dst), "v"(src) : "memory");
}

__device__ __forceinline__ void wait_async0() {
  asm volatile("s_wait_asynccnt 0" ::: "memory");
}

// ---------------------------------------------------------------------------
// AdderNet SAD conv: out[n,c,p] = -sum_k |patch[n,p,k] - w[c,k]|
// Block: 256 threads -> 64 pixels x 64 couts, 4x4 register tile per thread.
// Double-buffered async global->LDS staging: chunk i+1's DMA overlaps chunk
// i's SAD compute. OOB/padding lanes redirect to a zeroed scratch line so
// the loader is fully branchless.
// K is a multiple of 32 for all six layers: 512, 2304, 512, 1152, 256, 1152.
// ---------------------------------------------------------------------------
template<int KH>
__global__ __launch_bounds__(256)
void sad_conv(const float* __restrict__ x, const float* __restrict__ w,
              float* __restrict__ out, const float* __restrict__ zb,
              int N, int Cin, int H, int W, int Cout,
              int Ho, int Wo, int stride, int pad, int tilesPerN)
{
  __shared__ float As[2][32 * 64];   // patches, k-major [k][pixel]
  __shared__ float Bs[2][32 * 64];   // weights, k-major [k][cout]

  const int tid  = threadIdx.x;
  const int tx   = tid & 15;       // pixel group (4 pixels)
  const int ty   = tid >> 4;       // cout group  (4 couts)
  const int n    = blockIdx.x / tilesPerN;
  const int pix0 = (blockIdx.x % tilesPerN) * 64;
  const int c0   = blockIdx.y * 64;
  const int HW   = H * W;
  const int HoWo = Ho * Wo;
  const int K    = Cin * KH * KH;

  // ---- loop-invariant loader state (hoisted out of the K loop) ----
  const int klB  = tid & 31;            // B k-row owned by this thread
  const int clB0 = tid >> 5;            // B col = e*8 + clB0
  const int plA  = tid & 63;            // A pixel col owned by this thread
  const int klA0 = tid >> 6;            // A k-row = e*4 + klA0
  const int pixA = pix0 + plA;
  const bool pixOK = pixA < HoWo;
  const float* zbp = zb + plA;          // zero fallback (64-float line)

  const float* wBase = w + (size_t)c0 * K + klB;          // + cl*K + k0
  const float* xBase = x + (size_t)n * Cin * HW;          // image base

  int oyS = 0, oxS = 0;
  if (KH == 3) {
    int oy = pixA / Wo;                 // once per thread, not per chunk
    int ox = pixA - oy * Wo;
    oyS = oy * stride - pad;
    oxS = ox * stride - pad;
  }
  const float* x1Base = pixOK ? xBase + pixA : zbp;       // 1x1 base
  const size_t hwStep = pixOK ? (size_t)HW : 0;           // 1x1 k-stride

  uint32_t ldsB0[8], ldsA0[8];
#pragma unroll
  for (int e = 0; e < 8; ++e) {
    ldsB0[e] = lds_off(&Bs[0][klB * 64 + (e * 8 + clB0)]);
    ldsA0[e] = lds_off(&As[0][(e * 4 + klA0) * 64 + plA]);
  }

  // issue one 32-wide K-chunk of async DMA into buffer `sel`
  auto issue_chunk = [&](int k0, int sel) {
    const uint32_t boff = (uint32_t)sel * (32u * 64u * 4u);
#pragma unroll
    for (int e = 0; e < 8; ++e) {
      const float* g = wBase + (size_t)(e * 8 + clB0) * K + k0;
      async_ld_f32(ldsB0[e] + boff, g);
    }
    if (KH == 1) {
#pragma unroll
      for (int e = 0; e < 8; ++e) {
        const int kl = e * 4 + klA0;
        const float* g = x1Base + (size_t)(k0 + kl) * hwStep;
        async_ld_f32(ldsA0[e] + boff, g);
      }
    } else {
#pragma unroll
      for (int e = 0; e < 8; ++e) {
        const int k  = k0 + e * 4 + klA0;
        const int ci = k / 9;           // constant divisor -> magic mul
        const int r  = k - ci * 9;
        const int dy = r / 3;
        const int dx = r - dy * 3;
        const int iy = oyS + dy;
        const int ix = oxS + dx;
        const bool ok = pixOK & (iy >= 0) & (iy < H) & (ix >= 0) & (ix < W);
        const float* g = ok
            ? x + ((size_t)(n * Cin + ci) * HW + iy * W + ix)
            : zbp;
        async_ld_f32(ldsA0[e] + boff, g);
      }
    }
  };

  float acc[4][4];
#pragma unroll
  for (int i = 0; i < 4; ++i)
#pragma unroll
    for (int j = 0; j < 4; ++j) acc[i][j] = 0.f;

  const int nChunks = K >> 5;
  issue_chunk(0, 0);

  for (int i = 0; i < nChunks; ++i) {
    wait_async0();      // this wave's chunk-i LDS writes landed
    __syncthreads();    // all waves' writes visible; all done reading buf^1

    if (i + 1 < nChunks)                 // overlap next chunk's DMA
      issue_chunk((i + 1) << 5, (i + 1) & 1);

    const float* Ab = As[i & 1];
    const float* Bb = Bs[i & 1];

    // ---- SAD accumulate: 2 ds_load_b128 + 32 VALU per k-step ----
#pragma unroll 8
    for (int kl = 0; kl < 32; ++kl) {
      const float4 av = *(const float4*)(Ab + kl * 64 + (tx << 2));
      const float4 bv = *(const float4*)(Bb + kl * 64 + (ty << 2));
      const float a[4] = {av.x, av.y, av.z, av.w};
      const float b[4] = {bv.x, bv.y, bv.z, bv.w};
#pragma unroll
      for (int ii = 0; ii < 4; ++ii)
#pragma unroll
        for (int jj = 0; jj < 4; ++jj)
          acc[ii][jj] += fabsf(a[ii] - b[jj]);
    }
  }

  // ---- store -SAD (vectorized along contiguous pixels) ----
  const int pixBase = pix0 + (tx << 2);
  const bool fullRow = (pixBase + 3) < HoWo;
#pragma unroll
  for (int j = 0; j < 4; ++j) {
    const int c = c0 + (ty << 2) + j;
    const size_t row = ((size_t)n * Cout + c) * HoWo;
    if (fullRow) {
      float4 v;
      v.x = -acc[0][j]; v.y = -acc[1][j]; v.z = -acc[2][j]; v.w = -acc[3][j];
      *(float4*)(out + row + pixBase) = v;
    } else {
#pragma unroll
      for (int i = 0; i < 4; ++i) {
        const int pix = pixBase + i;
        if (pix < HoWo) out[row + pix] = -acc[i][j];
      }
    }
  }
}

// ---------------------------------------------------------------------------
// BN statistics via WMMA ones-reduction: D = A x ones + C accumulates
// per-channel sums (and sums of squares) exactly in f32 on the matrix unit.
// A is 16 channels x 4 pixels per wave-step (V_WMMA_F32_16X16X4_F32 layout:
// lanes 0-15 hold K=0,1; lanes 16-31 hold K=2,3; M = lane % 16).
// Wave id goes through readfirstlane so the trip count is scalar-uniform:
// no EXEC-mask loop machinery, EXEC stays all-ones for the WMMAs.
// ---------------------------------------------------------------------------
__global__ __launch_bounds__(256)
void bn_stats_wmma(const float* __restrict__ x, float* __restrict__ S,
                   float* __restrict__ SS, const float* __restrict__ zb,
                   int N, int C, int HW)
{
  const int c0   = blockIdx.x * 16;
  const int lane = threadIdx.x & 31;
  const int gw   = __builtin_amdgcn_readfirstlane(
                     blockIdx.y * (blockDim.x >> 5) + (threadIdx.x >> 5));
  const int numW = gridDim.y * (blockDim.x >> 5);
  const int m    = lane & 15;
  const int kb   = (lane >> 4) << 1;   // 0 or 2
  const float* zbp = zb + lane;

  v8f d  = {0.f, 0.f, 0.f, 0.f, 0.f, 0.f, 0.f, 0.f};
  v8f d2 = {0.f, 0.f, 0.f, 0.f, 0.f, 0.f, 0.f, 0.f};
  const v2f ones = {1.f, 1.f};

  for (int nn = 0; nn < N; ++nn) {
    const float* base = x + ((size_t)nn * C + c0 + m) * HW;
    for (int p0 = gw * 4; p0 < HW; p0 += numW * 4) {
      const int pa = p0 + kb;
      const int pb = pa + 1;
      const float* ga = (pa < HW) ? base + pa : zbp;
      const float* gb = (pb < HW) ? base + pb : zbp;
      v2f a;
      a.x = *ga;
      a.y = *gb;
      v2f a2 = a * a;
      d  = __builtin_amdgcn_wmma_f32_16x16x4_f32(false, a,  false, ones,
                                                 (short)0, d,  false, false);
      d2 = __builtin_amdgcn_wmma_f32_16x16x4_f32(false, a2, false, ones,
                                                 (short)0, d2, false, false);
    }
  }

  // D column 0: lane 0 holds M=0..7 across the 8 acc VGPRs, lane 16 M=8..15.
  if (lane == 0 || lane == 16) {
    const int mb = (lane == 0) ? 0 : 8;
#pragma unroll
    for (int i = 0; i < 8; ++i) {
      atomicAdd(&S[c0 + mb + i],  d[i]);
      atomicAdd(&SS[c0 + mb + i], d2[i]);
    }
  }
}

// ---------------------------------------------------------------------------
// BN apply (training-mode batch stats, biased var) + ReLU6, in-place capable
// ---------------------------------------------------------------------------
__global__ __launch_bounds__(256)
void bn_apply(const float* __restrict__ xin, float* __restrict__ yout,
              const float* __restrict__ S, const float* __restrict__ SS,
              const float* __restrict__ g, const float* __restrict__ b,
              int C, int HW, float invCnt)
{
  const int p = blockIdx.x * blockDim.x + threadIdx.x;
  if (p >= HW) return;
  const int c = blockIdx.y;
  const int n = blockIdx.z;
  const size_t idx = ((size_t)n * C + c) * HW + p;
  const float mean = S[c] * invCnt;
  const float var  = SS[c] * invCnt - mean * mean;
  float y = (xin[idx] - mean) * rsqrtf(var + BN_EPS) * g[c] + b[c];
  y = fminf(fmaxf(y, 0.f), 6.f);
  yout[idx] = y;
}

// ---------------------------------------------------------------------------
// Host side
// ---------------------------------------------------------------------------
struct Cfg { int Cin, H, W, Cout, kh, stride, pad; };

static void run_layer(const float* in, const float* w, const float* g,
                      const float* b, float* convOut, float* bnOut,
                      float* stats, float* zb, int N, Cfg c,
                      hipStream_t stream)
{
  const int Ho = (c.H + 2 * c.pad - c.kh) / c.stride + 1;
  const int Wo = (c.W + 2 * c.pad - c.kh) / c.stride + 1;
  const int HoWo = Ho * Wo;
  const int tiles = (HoWo + 63) / 64;
  dim3 grid(tiles * N, c.Cout / 64);
  if (c.kh == 1)
    sad_conv<1><<<grid, 256, 0, stream>>>(in, w, convOut, zb, N, c.Cin, c.H,
                                          c.W, c.Cout, Ho, Wo, c.stride,
                                          c.pad, tiles);
  else
    sad_conv<3><<<grid, 256, 0, stream>>>(in, w, convOut, zb, N, c.Cin, c.H,
                                          c.W, c.Cout, Ho, Wo, c.stride,
                                          c.pad, tiles);

  hipMemsetAsync(stats, 0, 2 * c.Cout * sizeof(float), stream);
  dim3 sg(c.Cout / 16, 16);
  bn_stats_wmma<<<sg, 256, 0, stream>>>(convOut, stats, stats + c.Cout, zb,
                                        N, c.Cout, HoWo);
  dim3 ag((HoWo + 255) / 256, c.Cout, N);
  bn_apply<<<ag, 256, 0, stream>>>(convOut, bnOut, stats, stats + c.Cout, g,
                                   b, c.Cout, HoWo,
                                   1.f / ((float)N * (float)HoWo));
}

extern "C" void kernel_launch(void* const* d_in, const int* in_sizes, int n_in,
                              void* d_out, int out_size, void* d_ws,
                              size_t ws_size, hipStream_t stream)
{
  (void)in_sizes; (void)n_in; (void)out_size;
  const float* x  = (const float*)d_in[0];
  const float* w1 = (const float*)d_in[1];
  const float* g1 = (const float*)d_in[2];
  const float* b1 = (const float*)d_in[3];
  const float* w2 = (const float*)d_in[4];
  const float* g2 = (const float*)d_in[5];
  const float* b2 = (const float*)d_in[6];
  const float* w3 = (const float*)d_in[7];
  const float* g3 = (const float*)d_in[8];
  const float* b3 = (const float*)d_in[9];
  const float* w4 = (const float*)d_in[10];
  const float* g4 = (const float*)d_in[11];
  const float* b4 = (const float*)d_in[12];
  const float* w5 = (const float*)d_in[13];
  const float* g5 = (const float*)d_in[14];
  const float* b5 = (const float*)d_in[15];
  const float* w6 = (const float*)d_in[16];
  const float* g6 = (const float*)d_in[17];
  const float* b6 = (const float*)d_in[18];

  // workspace layout (floats): zero line | stats | ping | pong
  float* zb    = (float*)d_ws;
  float* stats = zb + 64;
  float* bufA  = stats + 1024;                       // 16*256*38*38
  float* bufB  = bufA + (size_t)16 * 256 * 38 * 38;  // 16*512*19*19
  const size_t needF = 64 + 1024 + (size_t)16 * 256 * 38 * 38
                     + (size_t)16 * 512 * 19 * 19;
  if (ws_size < needF * sizeof(float)) return;

  hipMemsetAsync(zb, 0, 64 * sizeof(float), stream);

  const int N = 16;
  run_layer(x,    w1, g1, b1, bufA, bufA,           stats, zb, N,
            Cfg{512, 38, 38, 256, 1, 1, 0}, stream);
  run_layer(bufA, w2, g2, b2, bufB, bufB,           stats, zb, N,
            Cfg{256, 38, 38, 512, 3, 2, 1}, stream);
  run_layer(bufB, w3, g3, b3, bufA, bufA,           stats, zb, N,
            Cfg{512, 19, 19, 128, 1, 1, 0}, stream);
  run_layer(bufA, w4, g4, b4, bufB, bufB,           stats, zb, N,
            Cfg{128, 19, 19, 256, 3, 2, 1}, stream);
  run_layer(bufB, w5, g5, b5, bufA, bufA,           stats, zb, N,
            Cfg{256, 10, 10, 128, 1, 1, 0}, stream);
  run_layer(bufA, w6, g6, b6, bufB, (float*)d_out,  stats, zb, N,
            Cfg{128, 10, 10, 256, 3, 2, 0}, stream);
}